// CustomLoss_30743375905383
// MI455X (gfx1250) — compile-verified
//
#include <hip/hip_runtime.h>

// ---------------------------------------------------------------------------
// CustomLoss reduction for MI455X (gfx1250, wave32).
//
// Bandwidth-bound: 192 MB single-pass stream -> ~8.6 us floor @ 23.3 TB/s.
// Data path: coalesced b128 loads (float4 pred, int4 target) -> per-lane f32
// accumulator. Cross-lane reduction uses V_WMMA_F32_16X16X4_F32 with an
// all-ones B matrix (exact ones-weighted sum of the 32 lane partials).
// ---------------------------------------------------------------------------

typedef __attribute__((ext_vector_type(2))) float v2f;
typedef __attribute__((ext_vector_type(8))) float v8f;

// Reduce 32 per-lane f32 partials to their total, returned on every lane.
// A-matrix (16x4 f32, 2 VGPRs/lane): VGPR0 = partial, VGPR1 = 0
//   -> A[m][0] = s_{lane m}, A[m][2] = s_{lane 16+m}, A[m][1]=A[m][3]=0
// B = all ones (layout-independent) -> D[m][n] = s_m + s_{16+m}
// Each lane sums its 8 D VGPRs: lanes 0-15 get rows 0-7, lanes 16-31 rows
// 8-15; one shfl_xor(16) completes the sum. Requires EXEC == all ones.
__device__ __forceinline__ float wave_reduce32(float s) {
    v2f a; a[0] = s;    a[1] = 0.0f;
    v2f b; b[0] = 1.0f; b[1] = 1.0f;
    v8f c = {};
    v8f d = __builtin_amdgcn_wmma_f32_16x16x4_f32(
        /*neg_a=*/false, a, /*neg_b=*/false, b,
        /*c_mod=*/(short)0, c, /*reuse_a=*/false, /*reuse_b=*/false);
    float r = ((d[0] + d[1]) + (d[2] + d[3])) + ((d[4] + d[5]) + (d[6] + d[7]));
    r += __shfl_xor(r, 16, 32);
    return r;
}

// 256-thread block reduction; valid on threadIdx.x == 0.
__device__ __forceinline__ float block_reduce256(float s, float* lds) {
    float w = wave_reduce32(s);
    const int wave = threadIdx.x >> 5;
    const int lane = threadIdx.x & 31;
    if (lane == 0) lds[wave] = w;
    __syncthreads();
    float tot = 0.0f;
    if (threadIdx.x == 0) {
#pragma unroll
        for (int i = 0; i < 8; ++i) tot += lds[i];
    }
    return tot;
}

// Per-element loss term (branch-free select form of the reference).
__device__ __forceinline__ float loss_elem(float p0, float p1, int t) {
    const bool z = (t == 0);
    const float a = z ? (1.0f - p0) : p0;
    const float b = z ? p1 : (1.0f - p1);
    const bool wrong = z ? (p0 < p1) : (p1 < p0);
    return (a * a + b * b) + (wrong ? 2.0f : 0.0f);
}

__global__ void loss_partial_kernel(const float* __restrict__ pred,
                                    const int* __restrict__ tgt,
                                    float* __restrict__ partials,
                                    int n4, int n) {
    __shared__ float lds[8];
    const float4* __restrict__ pred4 = (const float4*)pred;
    const int4* __restrict__ tgt4 = (const int4*)tgt;

    const int gtid = blockIdx.x * blockDim.x + threadIdx.x;
    const int stride = gridDim.x * blockDim.x;

    float acc = 0.0f;
    // 4 rows per iteration: two float4 pred loads + one int4 target load.
    for (int i = gtid; i < n4; i += stride) {
        const float4 pa = pred4[2 * i];
        const float4 pb = pred4[2 * i + 1];
        const int4 t = tgt4[i];
        acc += loss_elem(pa.x, pa.y, t.x);
        acc += loss_elem(pa.z, pa.w, t.y);
        acc += loss_elem(pb.x, pb.y, t.z);
        acc += loss_elem(pb.z, pb.w, t.w);
    }
    // Scalar tail (0..3 rows); re-converges before the WMMA reduction.
    const int rem = n - 4 * n4;
    if (gtid < rem) {
        const int j = 4 * n4 + gtid;
        acc += loss_elem(pred[2 * j], pred[2 * j + 1], tgt[j]);
    }

    const float tot = block_reduce256(acc, lds);
    if (threadIdx.x == 0) partials[blockIdx.x] = tot;
}

__global__ void loss_final_kernel(const float* __restrict__ partials,
                                  float* __restrict__ out,
                                  int nblocks, int n) {
    __shared__ float lds[8];
    float s = 0.0f;
    // nblocks is a multiple of blockDim.x -> uniform trip count, no
    // divergence before the WMMA reduction.
    for (int i = threadIdx.x; i < nblocks; i += blockDim.x) s += partials[i];
    const float tot = block_reduce256(s, lds);
    if (threadIdx.x == 0) out[0] = tot / (float)n;
}

extern "C" void kernel_launch(void* const* d_in, const int* in_sizes, int n_in,
                              void* d_out, int out_size, void* d_ws, size_t ws_size,
                              hipStream_t stream) {
    const float* pred = (const float*)d_in[0]; // (N, 2) float32, row-major
    const int* tgt = (const int*)d_in[1];      // (N,) int
    const int n = in_sizes[1];
    const int n4 = n / 4;

    float* partials = (float*)d_ws; // 2048 floats = 8 KB scratch

    constexpr int THREADS = 256; // 8 waves (wave32)
    constexpr int BLOCKS = 2048; // ~64 waves/WGP of latency hiding; 8 iters/thread

    loss_partial_kernel<<<BLOCKS, THREADS, 0, stream>>>(pred, tgt, partials, n4, n);
    loss_final_kernel<<<1, THREADS, 0, stream>>>(partials, (float*)d_out, BLOCKS, n);
}